// EmotionalEmbeddingSpace_49667001811007
// MI455X (gfx1250) — compile-verified
//
#include <hip/hip_runtime.h>
#include <hip/hip_bf16.h>

typedef __attribute__((ext_vector_type(2))) float v2f;
typedef __attribute__((ext_vector_type(8))) float v8f;

#define DIM      768
#define HID      1536
#define BATCH    2048
#define TSTEPS   32
#define SUBSTEPS 4

// D(16x16,f32) += A(16x4,f32) * B(4x16,f32)  -- full f32 matrix pipe (CDNA5)
__device__ __forceinline__ v8f wmma_k4(v2f a, v2f b, v8f c) {
  return __builtin_amdgcn_wmma_f32_16x16x4_f32(false, a, false, b, (short)0, c,
                                               false, false);
}

// Pack W[K][N] row-major into pair-interleaved Wp[(K/2)][N][2] so that a WMMA
// B fragment (pair of adjacent-K values at one column) is a single b64 load.
__global__ void __launch_bounds__(256)
ee_pack_pairs(const float* __restrict__ W, float* __restrict__ Wp, int K,
              int N) {
  const int idx = blockIdx.x * blockDim.x + threadIdx.x;  // over (K/2)*N
  const int total = (K >> 1) * N;
  if (idx >= total) return;
  const int col = idx % N;
  const int kp = idx / N;
  v2f v = { W[(size_t)(2 * kp) * N + col], W[(size_t)(2 * kp + 1) * N + col] };
  *(v2f*)(Wp + (size_t)idx * 2) = v;
}

// H[BATCH,HID] = tanh(Yin[BATCH,DIM] @ W1[DIM,HID] + b1)
// one wave -> 16x64 output tile (4 accumulators share one A fragment)
template <bool PACKED>
__global__ void __launch_bounds__(256, 1)
ee_gemm1_tanh(const float* __restrict__ Yin, const float* __restrict__ W1,
              const float* __restrict__ b1, float* __restrict__ H) {
  const int lane = threadIdx.x & 31;
  const int wave = blockIdx.x * 8 + (threadIdx.x >> 5);
  const int rowTile = wave & 127;          // BATCH/16 = 128 row tiles
  const int colTile = wave >> 7;           // HID/64   = 24 col tiles
  const int l16 = lane & 15;
  const int half = lane >> 4;
  const int row0 = rowTile << 4;
  const int col0 = colTile << 6;

  v8f c0 = {}, c1 = {}, c2 = {}, c3 = {};
  const float* A = Yin + (size_t)(row0 + l16) * DIM + (half << 1);

  if constexpr (PACKED) {
    // pair layout: lane reads (W1[k+2h][col], W1[k+2h+1][col]) as one b64
    const float* B = W1 + ((size_t)half * HID + col0 + l16) * 2;
    for (int k = 0; k < DIM; k += 4) {
      v2f a = *(const v2f*)(A + k);
      const float* bp = B + (size_t)(k >> 1) * (2 * HID);
      v2f b0 = *(const v2f*)(bp);
      v2f bv1 = *(const v2f*)(bp + 32);
      v2f bv2 = *(const v2f*)(bp + 64);
      v2f bv3 = *(const v2f*)(bp + 96);
      c0 = wmma_k4(a, b0, c0);
      c1 = wmma_k4(a, bv1, c1);
      c2 = wmma_k4(a, bv2, c2);
      c3 = wmma_k4(a, bv3, c3);
    }
  } else {
    const float* B = W1 + (size_t)(half << 1) * HID + col0 + l16;
    for (int k = 0; k < DIM; k += 4) {
      v2f a = *(const v2f*)(A + k);
      const float* bp = B + (size_t)k * HID;
      v2f b0 = { bp[0], bp[HID] };
      v2f bv1 = { bp[16], bp[HID + 16] };
      v2f bv2 = { bp[32], bp[HID + 32] };
      v2f bv3 = { bp[48], bp[HID + 48] };
      c0 = wmma_k4(a, b0, c0);
      c1 = wmma_k4(a, bv1, c1);
      c2 = wmma_k4(a, bv2, c2);
      c3 = wmma_k4(a, bv3, c3);
    }
  }

  const float bb0 = b1[col0 + l16];
  const float bb1 = b1[col0 + 16 + l16];
  const float bb2 = b1[col0 + 32 + l16];
  const float bb3 = b1[col0 + 48 + l16];
#pragma unroll
  for (int j = 0; j < 8; ++j) {
    const int row = row0 + j + (half << 3);  // C layout: VGPR j -> M=j / M=8+j
    float* hp = H + (size_t)row * HID + col0 + l16;
    hp[0] = tanhf(c0[j] + bb0);
    hp[16] = tanhf(c1[j] + bb1);
    hp[32] = tanhf(c2[j] + bb2);
    hp[48] = tanhf(c3[j] + bb3);
  }
}

// kval = (H@W2 + b2) + tanh(Yin@Wm + bm), fused RK4 stage update in epilogue.
template <bool PACKED>
__global__ void __launch_bounds__(256, 1)
ee_gemm2_stage(const float* __restrict__ H, const float* __restrict__ W2,
               const float* __restrict__ b2, const float* __restrict__ Yin,
               const float* __restrict__ Wm, const float* __restrict__ bm,
               float* __restrict__ y, float* __restrict__ acc,
               float* __restrict__ ytmp, const float* __restrict__ tspan,
               int interval, int stage) {
  const int lane = threadIdx.x & 31;
  const int wave = blockIdx.x * 8 + (threadIdx.x >> 5);
  const int rowTile = wave & 127;          // 128 row tiles
  const int colTile = wave >> 7;           // DIM/64 = 12 col tiles
  const int l16 = lane & 15;
  const int half = lane >> 4;
  const int row0 = rowTile << 4;
  const int col0 = colTile << 6;

  // --- dynamics GEMM: H[BATCH,HID] @ W2[HID,DIM] ---
  v8f d0 = {}, d1 = {}, d2 = {}, d3 = {};
  {
    const float* A = H + (size_t)(row0 + l16) * HID + (half << 1);
    if constexpr (PACKED) {
      const float* B = W2 + ((size_t)half * DIM + col0 + l16) * 2;
      for (int k = 0; k < HID; k += 4) {
        v2f a = *(const v2f*)(A + k);
        const float* bp = B + (size_t)(k >> 1) * (2 * DIM);
        v2f b0 = *(const v2f*)(bp);
        v2f bv1 = *(const v2f*)(bp + 32);
        v2f bv2 = *(const v2f*)(bp + 64);
        v2f bv3 = *(const v2f*)(bp + 96);
        d0 = wmma_k4(a, b0, d0);
        d1 = wmma_k4(a, bv1, d1);
        d2 = wmma_k4(a, bv2, d2);
        d3 = wmma_k4(a, bv3, d3);
      }
    } else {
      const float* B = W2 + (size_t)(half << 1) * DIM + col0 + l16;
      for (int k = 0; k < HID; k += 4) {
        v2f a = *(const v2f*)(A + k);
        const float* bp = B + (size_t)k * DIM;
        v2f b0 = { bp[0], bp[DIM] };
        v2f bv1 = { bp[16], bp[DIM + 16] };
        v2f bv2 = { bp[32], bp[DIM + 32] };
        v2f bv3 = { bp[48], bp[DIM + 48] };
        d0 = wmma_k4(a, b0, d0);
        d1 = wmma_k4(a, bv1, d1);
        d2 = wmma_k4(a, bv2, d2);
        d3 = wmma_k4(a, bv3, d3);
      }
    }
  }
  // --- memory GEMM: Yin[BATCH,DIM] @ Wm[DIM,DIM] ---
  v8f m0 = {}, m1 = {}, m2 = {}, m3 = {};
  {
    const float* A = Yin + (size_t)(row0 + l16) * DIM + (half << 1);
    if constexpr (PACKED) {
      const float* B = Wm + ((size_t)half * DIM + col0 + l16) * 2;
      for (int k = 0; k < DIM; k += 4) {
        v2f a = *(const v2f*)(A + k);
        const float* bp = B + (size_t)(k >> 1) * (2 * DIM);
        v2f b0 = *(const v2f*)(bp);
        v2f bv1 = *(const v2f*)(bp + 32);
        v2f bv2 = *(const v2f*)(bp + 64);
        v2f bv3 = *(const v2f*)(bp + 96);
        m0 = wmma_k4(a, b0, m0);
        m1 = wmma_k4(a, bv1, m1);
        m2 = wmma_k4(a, bv2, m2);
        m3 = wmma_k4(a, bv3, m3);
      }
    } else {
      const float* B = Wm + (size_t)(half << 1) * DIM + col0 + l16;
      for (int k = 0; k < DIM; k += 4) {
        v2f a = *(const v2f*)(A + k);
        const float* bp = B + (size_t)k * DIM;
        v2f b0 = { bp[0], bp[DIM] };
        v2f bv1 = { bp[16], bp[DIM + 16] };
        v2f bv2 = { bp[32], bp[DIM + 32] };
        v2f bv3 = { bp[48], bp[DIM + 48] };
        m0 = wmma_k4(a, b0, m0);
        m1 = wmma_k4(a, bv1, m1);
        m2 = wmma_k4(a, bv2, m2);
        m3 = wmma_k4(a, bv3, m3);
      }
    }
  }

  const float hstep =
      (tspan[interval + 1] - tspan[interval]) * (1.0f / (float)SUBSTEPS);
  const float b2c0 = b2[col0 + l16], b2c1 = b2[col0 + 16 + l16],
              b2c2 = b2[col0 + 32 + l16], b2c3 = b2[col0 + 48 + l16];
  const float bmc0 = bm[col0 + l16], bmc1 = bm[col0 + 16 + l16],
              bmc2 = bm[col0 + 32 + l16], bmc3 = bm[col0 + 48 + l16];

#pragma unroll
  for (int j = 0; j < 8; ++j) {
    const int row = row0 + j + (half << 3);
    const size_t base = (size_t)row * DIM + col0 + l16;
    float kv0 = (d0[j] + b2c0) + tanhf(m0[j] + bmc0);
    float kv1 = (d1[j] + b2c1) + tanhf(m1[j] + bmc1);
    float kv2 = (d2[j] + b2c2) + tanhf(m2[j] + bmc2);
    float kv3 = (d3[j] + b2c3) + tanhf(m3[j] + bmc3);
    float kvals[4] = { kv0, kv1, kv2, kv3 };
#pragma unroll
    for (int t = 0; t < 4; ++t) {
      const size_t idx = base + (size_t)(t << 4);
      const float kval = kvals[t];
      if (stage == 0) {            // k1
        acc[idx] = kval;
        ytmp[idx] = y[idx] + 0.5f * hstep * kval;
      } else if (stage == 1) {     // k2
        acc[idx] = acc[idx] + 2.0f * kval;
        ytmp[idx] = y[idx] + 0.5f * hstep * kval;
      } else if (stage == 2) {     // k3
        acc[idx] = acc[idx] + 2.0f * kval;
        ytmp[idx] = y[idx] + hstep * kval;
      } else {                     // k4 + state advance
        y[idx] = y[idx] + (hstep * (1.0f / 6.0f)) * (acc[idx] + kval);
      }
    }
  }
}

extern "C" void kernel_launch(void* const* d_in, const int* in_sizes, int n_in,
                              void* d_out, int out_size, void* d_ws,
                              size_t ws_size, hipStream_t stream) {
  const float* x = (const float*)d_in[0];
  const float* ts = (const float*)d_in[1];
  const float* W1 = (const float*)d_in[2];
  const float* b1 = (const float*)d_in[3];
  const float* W2 = (const float*)d_in[4];
  const float* b2 = (const float*)d_in[5];
  const float* Wm = (const float*)d_in[6];
  const float* bm = (const float*)d_in[7];
  float* out = (float*)d_out;

  // workspace layout: y | ytmp | acc | H [| W1p | W2p | Wmp]
  const size_t stateElems = (size_t)BATCH * DIM;
  float* y = (float*)d_ws;
  float* ytmp = y + stateElems;
  float* acc = ytmp + stateElems;
  float* Hbuf = acc + stateElems;
  float* W1p = Hbuf + (size_t)BATCH * HID;
  float* W2p = W1p + (size_t)DIM * HID;
  float* Wmp = W2p + (size_t)HID * DIM;

  const size_t needPacked =
      (3 * stateElems + (size_t)BATCH * HID + 2 * (size_t)DIM * HID +
       (size_t)DIM * DIM) *
      sizeof(float);
  const bool packed = (ws_size >= needPacked);

  const size_t stateBytes = stateElems * sizeof(float);
  hipMemcpyAsync(y, x, stateBytes, hipMemcpyDeviceToDevice, stream);
  hipMemcpyAsync(out, x, stateBytes, hipMemcpyDeviceToDevice, stream);

  const dim3 blk(256);
  const dim3 gridA((BATCH / 16) * (HID / 64) / 8);  // 384 blocks, 8 waves each
  const dim3 gridB((BATCH / 16) * (DIM / 64) / 8);  // 192 blocks

  const float* W1u = W1;
  const float* W2u = W2;
  const float* Wmu = Wm;
  if (packed) {
    const int t1 = (DIM / 2) * HID;
    const int t2 = (HID / 2) * DIM;
    const int t3 = (DIM / 2) * DIM;
    ee_pack_pairs<<<(t1 + 255) / 256, blk, 0, stream>>>(W1, W1p, DIM, HID);
    ee_pack_pairs<<<(t2 + 255) / 256, blk, 0, stream>>>(W2, W2p, HID, DIM);
    ee_pack_pairs<<<(t3 + 255) / 256, blk, 0, stream>>>(Wm, Wmp, DIM, DIM);
    W1u = W1p; W2u = W2p; Wmu = Wmp;
  }

  for (int iv = 0; iv < TSTEPS - 1; ++iv) {
    for (int s = 0; s < SUBSTEPS; ++s) {
      for (int stage = 0; stage < 4; ++stage) {
        const float* Yin = (stage == 0) ? y : ytmp;
        if (packed) {
          ee_gemm1_tanh<true><<<gridA, blk, 0, stream>>>(Yin, W1u, b1, Hbuf);
          ee_gemm2_stage<true><<<gridB, blk, 0, stream>>>(
              Hbuf, W2u, b2, Yin, Wmu, bm, y, acc, ytmp, ts, iv, stage);
        } else {
          ee_gemm1_tanh<false><<<gridA, blk, 0, stream>>>(Yin, W1u, b1, Hbuf);
          ee_gemm2_stage<false><<<gridB, blk, 0, stream>>>(
              Hbuf, W2u, b2, Yin, Wmu, bm, y, acc, ytmp, ts, iv, stage);
        }
      }
    }
    hipMemcpyAsync(out + (size_t)(iv + 1) * stateElems, y, stateBytes,
                   hipMemcpyDeviceToDevice, stream);
  }
}